// DynamicPPELoss_83829171684024
// MI455X (gfx1250) — compile-verified
//
#include <hip/hip_runtime.h>
#include <hip/hip_bf16.h>

#define N_PRIORS 8096
#define BATCH 128
#define N_OBJ 32
#define N_CLASSES 7
#define THRESHOLD 0.5f
#define PPB 256                                  // priors per block
#define CHUNKS ((N_PRIORS + PPB - 1) / PPB)      // 32
#define NBLK (BATCH * CHUNKS)                    // 4096 partial slots

typedef __attribute__((ext_vector_type(2))) float v2f;
typedef __attribute__((ext_vector_type(8))) float v8f;

// Full-precision 32-lane sum on the matrix pipe:
// A(16x4) rows = {v_m, 0, v_{m+16}, 0}, B = ones  =>  D[m][n] = v_m + v_{m+16}.
// Each lane sums its 8 D VGPRs (rows 0..7 or 8..15), then xor-16 shuffle
// combines the two halves. Every lane ends up with the full 32-lane total.
__device__ __forceinline__ float wave_sum32(float v) {
  v2f a; a[0] = v;    a[1] = 0.0f;
  v2f b; b[0] = 1.0f; b[1] = 1.0f;
  v8f c = {};
  v8f d = __builtin_amdgcn_wmma_f32_16x16x4_f32(
      /*neg_a=*/false, a, /*neg_b=*/false, b,
      /*c_mod=*/(short)0, c, /*reuse_a=*/false, /*reuse_b=*/false);
  float s = d[0] + d[1] + d[2] + d[3] + d[4] + d[5] + d[6] + d[7];
  s += __shfl_xor(s, 16, 32);
  return s;
}

__device__ __forceinline__ float smooth_l1(float x) {
  float ax = fabsf(x);
  return ax < 1.0f ? 0.5f * ax * ax : ax - 0.5f;
}

__global__ __launch_bounds__(PPB) void ssd_loss_partial(
    const float* __restrict__ locs,    // [B, P, 4]
    const float* __restrict__ scores,  // [B, P, 7]
    const float* __restrict__ boxes,   // [B, 32, 4] (x1,y1,x2,y2)
    const float* __restrict__ priors,  // [P, 4] (cx,cy,w,h)
    const int*   __restrict__ labels,  // [B, 32]
    float*       __restrict__ partials) {
  const int b     = blockIdx.y;
  const int chunk = blockIdx.x;
  const int tid   = threadIdx.x;

  __shared__ float sbx[N_OBJ][5];   // x1,y1,x2,y2,area
  __shared__ int   slab[N_OBJ];
  __shared__ float sred[8][3];

  if (tid < N_OBJ) {
    const float* bp = boxes + ((size_t)b * N_OBJ + tid) * 4;
    float x1 = bp[0], y1 = bp[1], x2 = bp[2], y2 = bp[3];
    sbx[tid][0] = x1; sbx[tid][1] = y1;
    sbx[tid][2] = x2; sbx[tid][3] = y2;
    sbx[tid][4] = (x2 - x1) * (y2 - y1);
    slab[tid] = labels[b * N_OBJ + tid];
  }
  __syncthreads();

  const int p = chunk * PPB + tid;
  float loc_sum = 0.0f, npos = 0.0f, nll = 0.0f;

  if (p < N_PRIORS) {
    const float* pl = locs   + ((size_t)b * N_PRIORS + p) * 4;
    const float* ps = scores + ((size_t)b * N_PRIORS + p) * N_CLASSES;
    // Overlap HBM latency with the IoU loop (global_prefetch_b8).
    __builtin_prefetch(pl, 0, 0);
    __builtin_prefetch(ps, 0, 0);

    float4 pr  = *(const float4*)(priors + (size_t)p * 4);
    float px1 = pr.x - pr.z * 0.5f;
    float py1 = pr.y - pr.w * 0.5f;
    float px2 = pr.x + pr.z * 0.5f;
    float py2 = pr.y + pr.w * 0.5f;
    float area_p = (px2 - px1) * (py2 - py1);

    float best = -1.0f;
    int   bestj = 0;
#pragma unroll 4
    for (int o = 0; o < N_OBJ; ++o) {
      float bx1 = sbx[o][0], by1 = sbx[o][1];
      float bx2 = sbx[o][2], by2 = sbx[o][3];
      float iw = fminf(bx2, px2) - fmaxf(bx1, px1);
      float ih = fminf(by2, py2) - fmaxf(by1, py1);
      float inter = fmaxf(iw, 0.0f) * fmaxf(ih, 0.0f);
      float uni = sbx[o][4] + area_p - inter;
      float iou = (uni > 0.0f) ? inter / fmaxf(uni, 1e-12f) : 0.0f;
      if (iou > best) { best = iou; bestj = o; }  // strict > => first-index argmax
    }

    const int tc = (best < THRESHOLD) ? 0 : slab[bestj];

    // Encode matched box against this prior.
    float mx1 = sbx[bestj][0], my1 = sbx[bestj][1];
    float mx2 = sbx[bestj][2], my2 = sbx[bestj][3];
    float cx = (mx1 + mx2) * 0.5f, cy = (my1 + my2) * 0.5f;
    float sw = mx2 - mx1,          sh = my2 - my1;
    float t0 = (cx - pr.x) / pr.z;
    float t1 = (cy - pr.y) / pr.w;
    float t2 = logf(sw / pr.z + 1e-8f);
    float t3 = logf(sh / pr.w + 1e-8f);

    float4 plv = *(const float4*)pl;
    if (tc != 0) {
      npos = 1.0f;
      loc_sum = smooth_l1(plv.x - t0) + smooth_l1(plv.y - t1) +
                smooth_l1(plv.z - t2) + smooth_l1(plv.w - t3);
    }

    // log-softmax NLL over 7 classes.
    float s0 = ps[0], s1 = ps[1], s2 = ps[2], s3 = ps[3];
    float s4 = ps[4], s5 = ps[5], s6 = ps[6];
    float m = fmaxf(fmaxf(fmaxf(s0, s1), fmaxf(s2, s3)),
                    fmaxf(fmaxf(s4, s5), s6));
    float se = __expf(s0 - m) + __expf(s1 - m) + __expf(s2 - m) +
               __expf(s3 - m) + __expf(s4 - m) + __expf(s5 - m) +
               __expf(s6 - m);
    float lse = m + logf(se);
    float st = s0;
    st = (tc == 1) ? s1 : st;
    st = (tc == 2) ? s2 : st;
    st = (tc == 3) ? s3 : st;
    st = (tc == 4) ? s4 : st;
    st = (tc == 5) ? s5 : st;
    st = (tc == 6) ? s6 : st;
    nll = lse - st;
  }

  // Converged here -> EXEC all-1s: legal to run WMMA wave reductions.
  float ls = wave_sum32(loc_sum);
  float ns = wave_sum32(npos);
  float cs = wave_sum32(nll);

  const int lane = tid & 31, wave = tid >> 5;
  if (lane == 0) { sred[wave][0] = ls; sred[wave][1] = ns; sred[wave][2] = cs; }
  __syncthreads();
  if (tid == 0) {
    float L = 0.0f, N = 0.0f, C = 0.0f;
#pragma unroll
    for (int w = 0; w < 8; ++w) { L += sred[w][0]; N += sred[w][1]; C += sred[w][2]; }
    const int blk = b * gridDim.x + chunk;
    partials[blk]            = L;
    partials[NBLK + blk]     = N;
    partials[2 * NBLK + blk] = C;
  }
}

__global__ __launch_bounds__(32) void ssd_loss_finalize(
    const float* __restrict__ partials, float* __restrict__ out) {
  const int lane = threadIdx.x;
  float L = 0.0f, N = 0.0f, C = 0.0f;
  for (int i = lane; i < NBLK; i += 32) {
    L += partials[i];
    N += partials[NBLK + i];
    C += partials[2 * NBLK + i];
  }
  L = wave_sum32(L);
  N = wave_sum32(N);
  C = wave_sum32(C);
  if (lane == 0) {
    float conf = C / (float)(BATCH * N_PRIORS);
    float loc  = L / (fmaxf(N, 1.0f) * 4.0f);
    out[0] = conf + loc;
  }
}

extern "C" void kernel_launch(void* const* d_in, const int* in_sizes, int n_in,
                              void* d_out, int out_size, void* d_ws, size_t ws_size,
                              hipStream_t stream) {
  const float* locs   = (const float*)d_in[0];  // predicted_locs   [128,8096,4]
  const float* scores = (const float*)d_in[1];  // predicted_scores [128,8096,7]
  const float* boxes  = (const float*)d_in[2];  // boxes            [128,32,4]
  const float* priors = (const float*)d_in[3];  // priors_cxcy      [8096,4]
  const int*   labels = (const int*)d_in[4];    // labels           [128,32]
  float* out      = (float*)d_out;
  float* partials = (float*)d_ws;               // 3 * 4096 floats = 48 KB

  dim3 grid(CHUNKS, BATCH);
  ssd_loss_partial<<<grid, PPB, 0, stream>>>(locs, scores, boxes, priors, labels, partials);
  ssd_loss_finalize<<<1, 32, 0, stream>>>(partials, out);
}